// HCFDA_38388417692287
// MI455X (gfx1250) — compile-verified
//
#include <hip/hip_runtime.h>
#include <hip/hip_bf16.h>
#include <math.h>

// ---------------------------------------------------------------------------
// Problem constants (match reference)
// ---------------------------------------------------------------------------
#define BB 8
#define CC 256
#define HH 128
#define WW 128
#define HW (HH * WW)          // 16384
#define HIDDEN 16             // C / R

typedef __attribute__((ext_vector_type(2))) float v2f;
typedef __attribute__((ext_vector_type(4))) float v4f;
typedef __attribute__((ext_vector_type(8))) float v8f;

__device__ __forceinline__ float sigmoidf_(float v) {
    return 1.0f / (1.0f + __expf(-v));
}

// ---------------------------------------------------------------------------
// 1) m[c] = mean over rows o of dct_w[o, c]   (fold channel-mean into weights)
// ---------------------------------------------------------------------------
__global__ __launch_bounds__(CC) void hcfda_dct_mean(const float* __restrict__ dctw,
                                                     float* __restrict__ m) {
    const int c = threadIdx.x;
    float s = 0.0f;
    #pragma unroll 8
    for (int o = 0; o < CC; ++o) s += dctw[o * CC + c];
    m[c] = s * (1.0f / (float)CC);
}

// ---------------------------------------------------------------------------
// 2) temp[b,hw] = sum_c x[b,c,hw] * m[c]   (float2 per thread, coalesced)
//    grid: B*HW/2 threads -> 256 blocks of 256
// ---------------------------------------------------------------------------
__global__ __launch_bounds__(256) void hcfda_temp(const float* __restrict__ x,
                                                  const float* __restrict__ m,
                                                  float* __restrict__ temp) {
    const int g   = blockIdx.x * 256 + threadIdx.x;   // float2 index over B*HW/2
    const int b   = g / (HW / 2);
    const int hw2 = g - b * (HW / 2);
    const v2f* xp = (const v2f*)(x + (size_t)b * CC * HW) + hw2;
    v2f acc = {0.0f, 0.0f};
    for (int c = 0; c < CC; ++c) {
        const v2f v = xp[c * (HW / 2)];   // stride HW/2 float2 between channels
        const float w = m[c];             // block-uniform -> scalar load
        acc.x += w * v.x;
        acc.y += w * v.y;
    }
    ((v2f*)temp)[g] = acc;
}

// ---------------------------------------------------------------------------
// 3) Per-(b,c) mean / max over HW.  Writes the stacked SE input matrix
//    Y[16][256]: rows 0..7 = y_avg (batch 0..7), rows 8..15 = y_max.
//    grid: B*C = 2048 blocks of 256 threads, float4 loads (x is L2-resident).
// ---------------------------------------------------------------------------
__global__ __launch_bounds__(256) void hcfda_reduce(const float* __restrict__ x,
                                                    float* __restrict__ Y) {
    const int bc = blockIdx.x;          // b*C + c
    const int b  = bc >> 8;
    const int c  = bc & 255;
    const v4f* p = (const v4f*)(x + (size_t)bc * HW);
    float s = 0.0f, mx = -INFINITY;
    #pragma unroll 4
    for (int i = 0; i < HW / (256 * 4); ++i) {
        const v4f v = p[threadIdx.x + i * 256];
        s  += (v.x + v.y) + (v.z + v.w);
        mx  = fmaxf(mx, fmaxf(fmaxf(v.x, v.y), fmaxf(v.z, v.w)));
    }
    #pragma unroll
    for (int off = 16; off > 0; off >>= 1) {
        s  += __shfl_xor(s, off, 32);
        mx  = fmaxf(mx, __shfl_xor(mx, off, 32));
    }
    __shared__ float ss[8], sm[8];
    const int wave = threadIdx.x >> 5;
    if ((threadIdx.x & 31) == 0) { ss[wave] = s; sm[wave] = mx; }
    __syncthreads();
    if (threadIdx.x == 0) {
        float S = 0.0f, M = -INFINITY;
        #pragma unroll
        for (int w = 0; w < 8; ++w) { S += ss[w]; M = fmaxf(M, sm[w]); }
        Y[b * CC + c]       = S * (1.0f / (float)HW);  // y_avg row
        Y[(8 + b) * CC + c] = M;                       // y_max row
    }
}

// ---------------------------------------------------------------------------
// 4) SE MLP on one wave with V_WMMA_F32_16X16X4_F32.
//    Z  = Y(16x256) @ w1^T(256x16)      -> 64 WMMA k-steps
//    H  = relu(Z + b1)                  -> LDS re-layout (C-layout -> A-layout)
//    O  = H(16x16)  @ w2^T(16x256)      -> 16 N-tiles x 4 k-steps
//    ca[b,o] = sigmoid(O[b,o]+b2) + sigmoid(O[8+b,o]+b2)   (shfl_xor 16)
//
//    A layout (16x4 f32): lane m=lane&15; VGPR0 = K (lane<16 ? 0 : 2),
//    VGPR1 = K+1.  C/D layout: VGPR r holds M = r (lanes 0-15) / r+8 (16-31),
//    N = lane&15.  (CDNA5 ISA §7.12.2)
// ---------------------------------------------------------------------------
__global__ __launch_bounds__(32) void hcfda_se_wmma(const float* __restrict__ Y,
                                                    const float* __restrict__ w1,
                                                    const float* __restrict__ b1,
                                                    const float* __restrict__ w2,
                                                    const float* __restrict__ b2,
                                                    float* __restrict__ ca) {
    __shared__ float Hs[16][16];
    const int lane  = threadIdx.x;
    const int nm    = lane & 15;              // M for A, N for B/C/D
    const int khalf = (lane >> 4) << 1;       // 0 (lanes 0-15) or 2 (16-31)
    const int mhi   = (lane >> 4) << 3;       // C/D row offset 0 or 8

    // ---- FC1: Z = Y @ w1^T ----
    v8f acc = {};
    for (int kb = 0; kb < CC; kb += 4) {
        v2f a, bm;
        a.x  = Y[nm * CC + kb + khalf];
        a.y  = Y[nm * CC + kb + khalf + 1];
        bm.x = w1[nm * CC + kb + khalf];      // B[k,n] = w1[n*256 + k]
        bm.y = w1[nm * CC + kb + khalf + 1];
        acc = __builtin_amdgcn_wmma_f32_16x16x4_f32(
                  false, a, false, bm, (short)0, acc, false, false);
    }
    const float bias1 = b1[nm];
    #pragma unroll
    for (int r = 0; r < 8; ++r) {
        float h = acc[r] + bias1;
        Hs[r + mhi][nm] = h > 0.0f ? h : 0.0f;   // relu, C-layout -> LDS
    }
    __syncthreads();

    // ---- FC2: O = H @ w2^T, 16 output tiles of N=16 ----
    for (int t = 0; t < 16; ++t) {
        v8f acc2 = {};
        #pragma unroll
        for (int ks = 0; ks < 4; ++ks) {
            v2f a, bm;
            a.x  = Hs[nm][4 * ks + khalf];
            a.y  = Hs[nm][4 * ks + khalf + 1];
            bm.x = w2[(16 * t + nm) * HIDDEN + 4 * ks + khalf];
            bm.y = w2[(16 * t + nm) * HIDDEN + 4 * ks + khalf + 1];
            acc2 = __builtin_amdgcn_wmma_f32_16x16x4_f32(
                       false, a, false, bm, (short)0, acc2, false, false);
        }
        const float bias2 = b2[16 * t + nm];
        #pragma unroll
        for (int r = 0; r < 8; ++r) {
            const float v     = sigmoidf_(acc2[r] + bias2);   // rows r / r+8
            const float other = __shfl_xor(v, 16, 32);        // avg <-> max half
            if (lane < 16) ca[r * CC + 16 * t + nm] = v + other;
        }
    }
}

// ---------------------------------------------------------------------------
// 5) One heat-diffusion step with reflect padding; optional sigmoid on last.
//    grid: B*HW threads = 512 blocks of 256.
// ---------------------------------------------------------------------------
__global__ __launch_bounds__(256) void hcfda_heat_step(const float* __restrict__ in,
                                                       float* __restrict__ out,
                                                       const float* __restrict__ lap,
                                                       const float* __restrict__ alphap,
                                                       int do_sigmoid) {
    const int g  = blockIdx.x * 256 + threadIdx.x;
    const int b  = g >> 14;            // /HW
    const int hw = g & (HW - 1);
    const int i  = hw >> 7;            // row
    const int j  = hw & (WW - 1);      // col
    const float* img = in + b * HW;
    const float alpha = *alphap;
    float acc = 0.0f;
    #pragma unroll
    for (int di = -1; di <= 1; ++di) {
        int r = i + di;
        r = (r < 0) ? -r : ((r > HH - 1) ? 2 * HH - 2 - r : r);
        #pragma unroll
        for (int dj = -1; dj <= 1; ++dj) {
            int q = j + dj;
            q = (q < 0) ? -q : ((q > WW - 1) ? 2 * WW - 2 - q : q);
            acc += lap[(di + 1) * 3 + (dj + 1)] * img[r * WW + q];
        }
    }
    float v = img[i * WW + j] + alpha * acc;
    if (do_sigmoid) v = sigmoidf_(v);
    out[g] = v;
}

// ---------------------------------------------------------------------------
// 6) out = x * sigmoid(ca[b,c] * heat[b,hw]); float4, non-temporal stores so
//    the 128MB output stream does not evict L2-resident x.
// ---------------------------------------------------------------------------
__global__ __launch_bounds__(256) void hcfda_final(const float* __restrict__ x,
                                                   const float* __restrict__ heat,
                                                   const float* __restrict__ ca,
                                                   float* __restrict__ out) {
    const int i4  = blockIdx.x * 256 + threadIdx.x;   // float4 index
    const int hw4 = i4 & (HW / 4 - 1);                // 4096 float4 per (b,c)
    const int c   = (i4 >> 12) & (CC - 1);
    const int b   = i4 >> 20;                         // / (C*HW/4)
    const float a = ca[(b << 8) + c];                 // block-uniform
    const v4f h   = ((const v4f*)heat)[(b << 12) + hw4];
    const v4f xv  = ((const v4f*)x)[i4];
    v4f o;
    o.x = xv.x * sigmoidf_(a * h.x);
    o.y = xv.y * sigmoidf_(a * h.y);
    o.z = xv.z * sigmoidf_(a * h.z);
    o.w = xv.w * sigmoidf_(a * h.w);
    __builtin_nontemporal_store(o, (v4f*)out + i4);
}

// ---------------------------------------------------------------------------
// Launch
// ---------------------------------------------------------------------------
extern "C" void kernel_launch(void* const* d_in, const int* in_sizes, int n_in,
                              void* d_out, int out_size, void* d_ws, size_t ws_size,
                              hipStream_t stream) {
    const float* x     = (const float*)d_in[0];
    const float* dct_w = (const float*)d_in[1];
    const float* w1    = (const float*)d_in[2];
    const float* b1    = (const float*)d_in[3];
    const float* w2    = (const float*)d_in[4];
    const float* b2    = (const float*)d_in[5];
    const float* alpha = (const float*)d_in[6];
    const float* lap   = (const float*)d_in[7];
    float* out = (float*)d_out;

    // workspace layout (floats)
    float* ws   = (float*)d_ws;
    float* m    = ws;                    // 256
    float* Y    = m + CC;                // 16*256 = 4096
    float* ca   = Y + 16 * CC;           // 8*256  = 2048
    float* temp = ca + BB * CC;          // B*HW   = 131072
    float* tA   = temp + BB * HW;        // B*HW   = 131072  (final heat lives here)

    hcfda_dct_mean<<<1, CC, 0, stream>>>(dct_w, m);
    hcfda_temp<<<(BB * HW / 2) / 256, 256, 0, stream>>>(x, m, temp);
    hcfda_reduce<<<BB * CC, 256, 0, stream>>>(x, Y);
    hcfda_se_wmma<<<1, 32, 0, stream>>>(Y, w1, b1, w2, b2, ca);

    // 3 diffusion steps: temp -> tA -> temp -> tA(sigmoid) == heat
    hcfda_heat_step<<<(BB * HW) / 256, 256, 0, stream>>>(temp, tA, lap, alpha, 0);
    hcfda_heat_step<<<(BB * HW) / 256, 256, 0, stream>>>(tA, temp, lap, alpha, 0);
    hcfda_heat_step<<<(BB * HW) / 256, 256, 0, stream>>>(temp, tA, lap, alpha, 1);

    hcfda_final<<<(BB * CC * HW / 4) / 256, 256, 0, stream>>>(x, tA, ca, out);
}